// DepthPredNet_35639638622598
// MI455X (gfx1250) — compile-verified
//
#include <hip/hip_runtime.h>
#include <math.h>

#define V_   5
#define B_   2
#define C_   32
#define D_   48
#define H_   128
#define W_   160
#define HW_  (H_*W_)        // 20480
#define DHW_ (D_*HW_)       // 983040
#define NSRC 4

typedef __attribute__((ext_vector_type(2))) float v2f;
typedef __attribute__((ext_vector_type(8))) float v8f;

// ---------------------------------------------------------------------------
// Standard 4x4 inverse (cofactor / gluInvertMatrix). Runs on 8 threads total.
// ---------------------------------------------------------------------------
__device__ inline void mat4_inv(const float m[16], float inv[16]) {
  inv[0]  =  m[5]*m[10]*m[15] - m[5]*m[11]*m[14] - m[9]*m[6]*m[15] + m[9]*m[7]*m[14] + m[13]*m[6]*m[11] - m[13]*m[7]*m[10];
  inv[4]  = -m[4]*m[10]*m[15] + m[4]*m[11]*m[14] + m[8]*m[6]*m[15] - m[8]*m[7]*m[14] - m[12]*m[6]*m[11] + m[12]*m[7]*m[10];
  inv[8]  =  m[4]*m[9]*m[15]  - m[4]*m[11]*m[13] - m[8]*m[5]*m[15] + m[8]*m[7]*m[13] + m[12]*m[5]*m[11] - m[12]*m[7]*m[9];
  inv[12] = -m[4]*m[9]*m[14]  + m[4]*m[10]*m[13] + m[8]*m[5]*m[14] - m[8]*m[6]*m[13] - m[12]*m[5]*m[10] + m[12]*m[6]*m[9];
  inv[1]  = -m[1]*m[10]*m[15] + m[1]*m[11]*m[14] + m[9]*m[2]*m[15] - m[9]*m[3]*m[14] - m[13]*m[2]*m[11] + m[13]*m[3]*m[10];
  inv[5]  =  m[0]*m[10]*m[15] - m[0]*m[11]*m[14] - m[8]*m[2]*m[15] + m[8]*m[3]*m[14] + m[12]*m[2]*m[11] - m[12]*m[3]*m[10];
  inv[9]  = -m[0]*m[9]*m[15]  + m[0]*m[11]*m[13] + m[8]*m[1]*m[15] - m[8]*m[3]*m[13] - m[12]*m[1]*m[11] + m[12]*m[3]*m[9];
  inv[13] =  m[0]*m[9]*m[14]  - m[0]*m[10]*m[13] - m[8]*m[1]*m[14] + m[8]*m[2]*m[13] + m[12]*m[1]*m[10] - m[12]*m[2]*m[9];
  inv[2]  =  m[1]*m[6]*m[15]  - m[1]*m[7]*m[14]  - m[5]*m[2]*m[15] + m[5]*m[3]*m[14] + m[13]*m[2]*m[7]  - m[13]*m[3]*m[6];
  inv[6]  = -m[0]*m[6]*m[15]  + m[0]*m[7]*m[14]  + m[4]*m[2]*m[15] - m[4]*m[3]*m[14] - m[12]*m[2]*m[7]  + m[12]*m[3]*m[6];
  inv[10] =  m[0]*m[5]*m[15]  - m[0]*m[7]*m[13]  - m[4]*m[1]*m[15] + m[4]*m[3]*m[13] + m[12]*m[1]*m[7]  - m[12]*m[3]*m[5];
  inv[14] = -m[0]*m[5]*m[14]  + m[0]*m[6]*m[13]  + m[4]*m[1]*m[14] - m[4]*m[2]*m[13] - m[12]*m[1]*m[6]  + m[12]*m[2]*m[5];
  inv[3]  = -m[1]*m[6]*m[11]  + m[1]*m[7]*m[10]  + m[5]*m[2]*m[11] - m[5]*m[3]*m[10] - m[9]*m[2]*m[7]   + m[9]*m[3]*m[6];
  inv[7]  =  m[0]*m[6]*m[11]  - m[0]*m[7]*m[10]  - m[4]*m[2]*m[11] + m[4]*m[3]*m[10] + m[8]*m[2]*m[7]   - m[8]*m[3]*m[6];
  inv[11] = -m[0]*m[5]*m[11]  + m[0]*m[7]*m[9]   + m[4]*m[1]*m[11] - m[4]*m[3]*m[9]  - m[8]*m[1]*m[7]   + m[8]*m[3]*m[5];
  inv[15] =  m[0]*m[5]*m[10]  - m[0]*m[6]*m[9]   - m[4]*m[1]*m[10] + m[4]*m[2]*m[9]  + m[8]*m[1]*m[6]   - m[8]*m[2]*m[5];
  float det = m[0]*inv[0] + m[1]*inv[4] + m[2]*inv[8] + m[3]*inv[12];
  float rd = 1.0f / det;
  #pragma unroll
  for (int k = 0; k < 16; ++k) inv[k] *= rd;
}

// K0: per (b, src-view) compute  full = src_new @ inv(ref_new); store rot(9)+tr(3)
__global__ void proj_setup_kernel(const float* __restrict__ proj, float* __restrict__ M) {
  int t = threadIdx.x;
  if (t >= B_ * NSRC) return;
  int b = t / NSRC, i = t % NSRC;
  float refm[16], srcm[16], refinv[16];
  // _combined_proj: rows0-2 = K[:3,:3] @ E[:3,:4], row3 = E row3
  for (int pass = 0; pass < 2; ++pass) {
    int v = (pass == 0) ? 0 : (i + 1);
    float* out = (pass == 0) ? refm : srcm;
    const float* E  = proj + ((size_t)(b * V_ + v) * 2 + 0) * 16;
    const float* Km = proj + ((size_t)(b * V_ + v) * 2 + 1) * 16;
    for (int r = 0; r < 3; ++r)
      for (int c = 0; c < 4; ++c) {
        float s = 0.f;
        for (int j = 0; j < 3; ++j) s += Km[r * 4 + j] * E[j * 4 + c];
        out[r * 4 + c] = s;
      }
    for (int c = 0; c < 4; ++c) out[12 + c] = E[12 + c];
  }
  mat4_inv(refm, refinv);
  float* o = M + t * 12;
  for (int r = 0; r < 3; ++r) {
    for (int c = 0; c < 3; ++c) {
      float s = 0.f;
      for (int j = 0; j < 4; ++j) s += srcm[r * 4 + j] * refinv[j * 4 + c];
      o[r * 3 + c] = s;
    }
  }
  for (int r = 0; r < 3; ++r) {
    float s = 0.f;
    for (int j = 0; j < 4; ++j) s += srcm[r * 4 + j] * refinv[j * 4 + 3];
    o[9 + r] = s;
  }
}

// K1: heavy warp + bilinear + aggregate. One thread per (b,d,h,w).
__global__ __launch_bounds__(256) void warp_agg_kernel(
    const float* __restrict__ feat, const float* __restrict__ dsamp,
    const float* __restrict__ vwt, const float* __restrict__ Mv,
    float* __restrict__ volOut) {
  const int hw = blockIdx.x * 256 + threadIdx.x;
  const int d  = blockIdx.y;
  const int b  = blockIdx.z;
  const int h  = hw / W_;
  const int w  = hw - h * W_;
  const float dep = dsamp[(size_t)(b * D_ + d) * HW_ + hw];
  const float fw = (float)w, fh = (float)h;

  float s[C_];
  #pragma unroll
  for (int c = 0; c < C_; ++c) s[c] = 0.f;
  float wsum = 1e-5f;

  for (int i = 0; i < NSRC; ++i) {
    const float* M = Mv + (b * NSRC + i) * 12;
    float X = (M[0] * fw + M[1] * fh + M[2]) * dep + M[9];
    float Y = (M[3] * fw + M[4] * fh + M[5]) * dep + M[10];
    float Z = (M[6] * fw + M[7] * fh + M[8]) * dep + M[11];
    bool valid = Z > 0.001f;
    float zs = valid ? Z : 1.f;
    float px = valid ? X / zs : -10000.f;
    float py = valid ? Y / zs : -10000.f;
    float x0 = floorf(px), y0 = floorf(py);
    float wx1 = px - x0, wx0 = 1.f - wx1;
    float wy1 = py - y0, wy0 = 1.f - wy1;
    // per-corner validity (matches reference: weight zeroed, index clamped)
    bool vx0 = (x0 >= 0.f)  && (x0 <= (float)(W_ - 1));
    bool vx1 = (x0 >= -1.f) && (x0 <= (float)(W_ - 2));
    bool vy0 = (y0 >= 0.f)  && (y0 <= (float)(H_ - 1));
    bool vy1 = (y0 >= -1.f) && (y0 <= (float)(H_ - 2));
    int ix0 = (int)fminf(fmaxf(x0,       0.f), (float)(W_ - 1));
    int ix1 = (int)fminf(fmaxf(x0 + 1.f, 0.f), (float)(W_ - 1));
    int iy0 = (int)fminf(fmaxf(y0,       0.f), (float)(H_ - 1));
    int iy1 = (int)fminf(fmaxf(y0 + 1.f, 0.f), (float)(H_ - 1));
    float wv = vwt[(size_t)(b * NSRC + i) * HW_ + hw];
    wsum += wv;
    float a00 = (vx0 && vy0) ? wx0 * wy0 * wv : 0.f;
    float a01 = (vx1 && vy0) ? wx1 * wy0 * wv : 0.f;
    float a10 = (vx0 && vy1) ? wx0 * wy1 * wv : 0.f;
    float a11 = (vx1 && vy1) ? wx1 * wy1 * wv : 0.f;
    int o00 = iy0 * W_ + ix0, o01 = iy0 * W_ + ix1;
    int o10 = iy1 * W_ + ix0, o11 = iy1 * W_ + ix1;
    const float* sf = feat + (size_t)((i + 1) * B_ + b) * C_ * HW_;
    #pragma unroll
    for (int c = 0; c < C_; ++c) {
      const float* p = sf + (size_t)c * HW_;
      s[c] += a00 * p[o00] + a01 * p[o01] + a10 * p[o10] + a11 * p[o11];
    }
  }
  const float invw = 1.f / wsum;
  const float* rf = feat + (size_t)b * C_ * HW_ + hw;   // view 0 = ref
  float* vo = volOut + (size_t)b * C_ * DHW_ + (size_t)d * HW_ + hw;
  #pragma unroll
  for (int c = 0; c < C_; ++c)
    vo[(size_t)c * DHW_] = rf[(size_t)c * HW_] * s[c] * invw;
}

// K2: prob_pre[b,d,h,w] = sum_c vol[b,c,d,h,w]*w_reg[c] + b_reg  via
// V_WMMA_F32_16X16X4_F32.  One wave handles 16 pixels (M), C in 8 K=4 chunks.
// A layout (ISA 7.12.2, 32-bit A 16x4): lanes 0-15 hold K=0,1; lanes 16-31 K=2,3.
// B columns all broadcast w_reg chunk -> every column of D holds the dot.
__global__ __launch_bounds__(256) void einsum_wmma_kernel(
    const float* __restrict__ vol, const float* __restrict__ wreg,
    const float* __restrict__ breg, float* __restrict__ pre) {
  const int tid  = blockIdx.x * 256 + threadIdx.x;
  const int wave = tid >> 5;
  const int lane = tid & 31;
  const int m    = lane & 15;   // pixel row within group
  const int kh   = lane >> 4;   // K half (0 -> K=0,1 ; 1 -> K=2,3)
  const long g0  = (long)wave * 16;            // first pixel of this group
  const int  b   = (int)(g0 / (long)DHW_);     // DHW_ % 16 == 0, no straddle
  const int  q0  = (int)(g0 - (long)b * DHW_);
  const int  q   = q0 + m;
  const float* vb = vol + (size_t)b * C_ * DHW_ + q;

  v8f acc = {0.f, 0.f, 0.f, 0.f, 0.f, 0.f, 0.f, 0.f};
  #pragma unroll
  for (int t = 0; t < 8; ++t) {
    const int c0 = 4 * t + 2 * kh;
    v2f a, bm;
    a[0]  = vb[(size_t)c0 * DHW_];
    a[1]  = vb[(size_t)(c0 + 1) * DHW_];
    bm[0] = wreg[c0];
    bm[1] = wreg[c0 + 1];
    acc = __builtin_amdgcn_wmma_f32_16x16x4_f32(
        /*neg_a=*/false, a, /*neg_b=*/false, bm,
        /*c_mod=*/(short)0, acc, /*reuse_a=*/false, /*reuse_b=*/false);
  }
  // D layout: VGPR r, lanes 0-15 -> M=r ; lanes 16-31 -> M=r+8. Columns identical.
  if (m == 0) {
    const float bb = breg[0];
    float* o = pre + (size_t)b * DHW_ + q0 + kh * 8;
    #pragma unroll
    for (int r = 0; r < 8; ++r) o[r] = acc[r] + bb;
  }
}

// K3: softmax over D + depth regression. Overwrites prob_pre with prob in place.
__global__ __launch_bounds__(256) void softmax_depth_kernel(
    const float* __restrict__ dsamp, float* __restrict__ prob,
    float* __restrict__ depth) {
  const int hw = blockIdx.x * 256 + threadIdx.x;
  const int b  = blockIdx.y;
  float pv[D_];
  float mx = -1e30f;
  #pragma unroll
  for (int d = 0; d < D_; ++d) {
    pv[d] = prob[(size_t)b * DHW_ + (size_t)d * HW_ + hw];
    mx = fmaxf(mx, pv[d]);
  }
  float ssum = 0.f;
  #pragma unroll
  for (int d = 0; d < D_; ++d) {
    pv[d] = expf(pv[d] - mx);
    ssum += pv[d];
  }
  const float r = 1.f / ssum;
  float dacc = 0.f;
  #pragma unroll
  for (int d = 0; d < D_; ++d) {
    float p = pv[d] * r;
    prob[(size_t)b * DHW_ + (size_t)d * HW_ + hw] = p;
    dacc += p * dsamp[(size_t)(b * D_ + d) * HW_ + hw];
  }
  depth[(size_t)b * HW_ + hw] = dacc;
}

extern "C" void kernel_launch(void* const* d_in, const int* in_sizes, int n_in,
                              void* d_out, int out_size, void* d_ws, size_t ws_size,
                              hipStream_t stream) {
  (void)in_sizes; (void)n_in; (void)out_size; (void)ws_size;
  const float* feat  = (const float*)d_in[0];  // (V,B,C,H,W)
  const float* proj  = (const float*)d_in[1];  // (B,V,2,4,4)
  const float* dsamp = (const float*)d_in[2];  // (B,D,H,W)
  const float* vwt   = (const float*)d_in[3];  // (B,4,H,W)
  const float* wreg  = (const float*)d_in[4];  // (C,)
  const float* breg  = (const float*)d_in[5];  // (1,)

  float* out    = (float*)d_out;
  float* depth  = out;                                   // B*H*W
  float* vw_out = depth + (size_t)B_ * HW_;              // B*4*H*W
  float* prob   = vw_out + (size_t)B_ * NSRC * HW_;      // B*D*H*W
  float* vol    = prob + (size_t)B_ * DHW_;              // B*C*D*H*W
  float* Mws    = (float*)d_ws;                          // 96 floats

  proj_setup_kernel<<<1, 32, 0, stream>>>(proj, Mws);

  hipMemcpyAsync(vw_out, vwt, (size_t)B_ * NSRC * HW_ * sizeof(float),
                 hipMemcpyDeviceToDevice, stream);

  dim3 g1(HW_ / 256, D_, B_);
  warp_agg_kernel<<<g1, 256, 0, stream>>>(feat, dsamp, vwt, Mws, vol);

  const int nwaves = (B_ * DHW_) / 16;                   // 122880 waves
  einsum_wmma_kernel<<<nwaves * 32 / 256, 256, 0, stream>>>(vol, wreg, breg, prob);

  dim3 g3(HW_ / 256, B_);
  softmax_depth_kernel<<<g3, 256, 0, stream>>>(dsamp, prob, depth);
}